// prop_58007828299964
// MI455X (gfx1250) — compile-verified
//
#include <hip/hip_runtime.h>
#include <hip/hip_bf16.h>
#include <cstddef>

#define BB     16
#define CIN    32
#define COUT   32
#define TT     13
#define NN     2000
#define KK     20
#define ALPHA  0.05f
#define NSLOPE 0.2f
#define CT     (CIN * TT)       // 416
#define DD     (TT * COUT)      // 416

typedef float v2f __attribute__((ext_vector_type(2)));
typedef float v8f __attribute__((ext_vector_type(8)));

__device__ __forceinline__ float lrelu(float v) {
    return v >= 0.0f ? v : NSLOPE * v;
}

// ---------------------------------------------------------------------------
// K1: u1[c*T+t] = sum_o a1[t*32+o] * W2[o,c]   (and u2 with a2), plus
//     c1 = sum_{t,o} a1[t*32+o]*b2[o], c2 likewise.  One block.
// ---------------------------------------------------------------------------
__global__ void k_uvec(const float* __restrict__ W2, const float* __restrict__ b2,
                       const float* __restrict__ a,
                       float* __restrict__ u1, float* __restrict__ u2,
                       float* __restrict__ c12) {
    int i = threadIdx.x;
    if (i < CT) {
        int c = i / TT, t = i % TT;
        float s1 = 0.f, s2 = 0.f;
        #pragma unroll
        for (int o = 0; o < COUT; ++o) {
            float w = W2[o * CIN + c];
            s1 += a[t * COUT + o] * w;
            s2 += a[DD + t * COUT + o] * w;
        }
        u1[i] = s1;
        u2[i] = s2;
    }
    if (i == 0) {
        float s1 = 0.f, s2 = 0.f;
        for (int t = 0; t < TT; ++t)
            for (int o = 0; o < COUT; ++o) {
                s1 += a[t * COUT + o] * b2[o];
                s2 += a[DD + t * COUT + o] * b2[o];
            }
        c12[0] = s1;
        c12[1] = s2;
    }
}

// ---------------------------------------------------------------------------
// K2: per adjacency row: relu, row-sum, normalize, iterative top-K argmax.
//     One 256-thread block per row. Lowest-index tie-break (matches jax).
// ---------------------------------------------------------------------------
__global__ void k_adj_topk(const float* __restrict__ adj,
                           int* __restrict__ idx, float* __restrict__ aval) {
    __shared__ float row[NN];
    __shared__ float sv[256];
    __shared__ int   si[256];
    const int n   = blockIdx.x;
    const int tid = threadIdx.x;

    float psum = 0.f;
    for (int j = tid; j < NN; j += 256) {
        float v = adj[(size_t)n * NN + j];
        v = v > 0.f ? v : 0.f;
        row[j] = v;
        psum += v;
    }
    sv[tid] = psum;
    __syncthreads();
    for (int s = 128; s > 0; s >>= 1) {
        if (tid < s) sv[tid] += sv[tid + s];
        __syncthreads();
    }
    const float inv = 1.0f / sv[0];
    __syncthreads();

    for (int k = 0; k < KK; ++k) {
        float bv = -2.0f;
        int   bj = 0x7fffffff;
        for (int j = tid; j < NN; j += 256) {
            float v = row[j];
            if (v > bv || (v == bv && j < bj)) { bv = v; bj = j; }
        }
        sv[tid] = bv; si[tid] = bj;
        __syncthreads();
        for (int s = 128; s > 0; s >>= 1) {
            if (tid < s) {
                float ov = sv[tid + s]; int oj = si[tid + s];
                if (ov > sv[tid] || (ov == sv[tid] && oj < si[tid])) {
                    sv[tid] = ov; si[tid] = oj;
                }
            }
            __syncthreads();
        }
        if (tid == 0) {
            int bestj = si[0];
            idx[n * KK + k]  = bestj;
            aval[n * KK + k] = sv[0] * inv;
            row[bestj] = -1.0f;   // exclude
        }
        __syncthreads();
    }
}

// ---------------------------------------------------------------------------
// K3: left[b,n] / right[b,n] = u{1,2} . x[b,:,:,n] + c{1,2}
// grid (ceil(N/256), B)
// ---------------------------------------------------------------------------
__global__ void k_leftright(const float* __restrict__ x,
                            const float* __restrict__ u1, const float* __restrict__ u2,
                            const float* __restrict__ c12,
                            float* __restrict__ left, float* __restrict__ right) {
    __shared__ float su1[CT], su2[CT];
    const int b = blockIdx.y;
    for (int i = threadIdx.x; i < CT; i += 256) { su1[i] = u1[i]; su2[i] = u2[i]; }
    __syncthreads();
    const int n = blockIdx.x * 256 + threadIdx.x;
    if (n < NN) {
        const float* xb = x + (size_t)b * CT * NN + n;
        float l = c12[0], r = c12[1];
        #pragma unroll 4
        for (int rI = 0; rI < CT; ++rI) {
            float xv = xb[(size_t)rI * NN];
            l += su1[rI] * xv;
            r += su2[rI] * xv;
        }
        left[b * NN + n]  = l;
        right[b * NN + n] = r;
    }
}

// ---------------------------------------------------------------------------
// K4: softmax denominator per (b,i); rmax[b] = max_j right[b,j].
// grid (ceil(N/256), B); right row staged in LDS.
// ---------------------------------------------------------------------------
__global__ void k_denom(const float* __restrict__ left, const float* __restrict__ right,
                        float* __restrict__ denom, float* __restrict__ rmax) {
    __shared__ float sr[NN];
    __shared__ float sred[256];
    const int b = blockIdx.y;
    const int tid = threadIdx.x;

    float lm = -3.4e38f;
    for (int j = tid; j < NN; j += 256) {
        float v = right[b * NN + j];
        sr[j] = v;
        lm = fmaxf(lm, v);
    }
    sred[tid] = lm;
    __syncthreads();
    for (int s = 128; s > 0; s >>= 1) {
        if (tid < s) sred[tid] = fmaxf(sred[tid], sred[tid + s]);
        __syncthreads();
    }
    const float rm = sred[0];
    if (blockIdx.x == 0 && tid == 0) rmax[b] = rm;

    const int i = blockIdx.x * 256 + tid;
    if (i < NN) {
        const float l  = left[b * NN + i];
        const float mm = lrelu(l + rm);   // true row max (lrelu is monotone)
        float s = 0.f;
        #pragma unroll 4
        for (int j = 0; j < NN; ++j)
            s += __expf(lrelu(l + sr[j]) - mm);
        denom[b * NN + i] = s;
    }
}

// ---------------------------------------------------------------------------
// K5: Aval[b,m,k] = attention(b, m, idx[m,k]) + adj1[m, idx[m,k]]
// ---------------------------------------------------------------------------
__global__ void k_attn(const int* __restrict__ idx, const float* __restrict__ aval,
                       const float* __restrict__ left, const float* __restrict__ right,
                       const float* __restrict__ denom, const float* __restrict__ rmax,
                       float* __restrict__ Aval) {
    const int gi = blockIdx.x * 256 + threadIdx.x;
    if (gi >= BB * NN * KK) return;
    const int k = gi % KK;
    const int m = (gi / KK) % NN;
    const int b = gi / (KK * NN);
    const int j = idx[m * KK + k];
    const float l  = left[b * NN + m];
    const float mm = lrelu(l + rmax[b]);
    const float attn = __expf(lrelu(l + right[b * NN + j]) - mm) / denom[b * NN + m];
    Aval[gi] = attn + aval[m * KK + k];
}

// ---------------------------------------------------------------------------
// K6: h1 = ALPHA*x + (1-ALPHA) * (x @ A^T)   (K-sparse gather)
// grid (N, B); block owns output column m of batch b.
// ---------------------------------------------------------------------------
__global__ void k_prop1(const float* __restrict__ x,
                        const int* __restrict__ idx, const float* __restrict__ Aval,
                        float* __restrict__ h1) {
    __shared__ int   sidx[KK];
    __shared__ float sav[KK];
    const int m = blockIdx.x, b = blockIdx.y, tid = threadIdx.x;
    if (tid < KK) {
        sidx[tid] = idx[m * KK + tid];
        sav[tid]  = Aval[((size_t)b * NN + m) * KK + tid];
    }
    __syncthreads();
    for (int r = tid; r < CT; r += 256) {
        const float* xr = x + ((size_t)b * CT + r) * NN;
        float acc = 0.f;
        #pragma unroll
        for (int k = 0; k < KK; ++k)
            acc += sav[k] * xr[sidx[k]];
        h1[((size_t)b * CT + r) * NN + m] = ALPHA * xr[m] + (1.0f - ALPHA) * acc;
    }
}

// ---------------------------------------------------------------------------
// K7: fused  h2 = ALPHA*x + (1-ALPHA)*(h1 @ A^T)   then  ho = W1 @ h2 + b1
// Block: (b, 16 columns m). Phase A builds h2 tile [CT][16] in LDS via gather.
// Phase B: per t, two 16x16 output tiles via V_WMMA_F32_16X16X4_F32, K=32 in
// 8 steps. 128 threads = 4 waves; wave w handles t = w, w+4, ...
// ---------------------------------------------------------------------------
__global__ void __launch_bounds__(128)
k_prop2_proj(const float* __restrict__ x, const float* __restrict__ h1,
             const int* __restrict__ idx, const float* __restrict__ Aval,
             const float* __restrict__ W1, const float* __restrict__ b1,
             float* __restrict__ out) {
    __shared__ float h2s[CT * 16];       // [r = c*T + t][mm]
    __shared__ int   sidx2[16 * KK];
    __shared__ float sav2[16 * KK];

    const int b   = blockIdx.y;
    const int m0  = blockIdx.x * 16;
    const int tid = threadIdx.x;

    // stage per-column sparse weights
    for (int i = tid; i < 16 * KK; i += 128) {
        const int mm = i / KK, k = i % KK;
        sidx2[i] = idx[(m0 + mm) * KK + k];
        sav2[i]  = Aval[((size_t)b * NN + m0 + mm) * KK + k];
    }
    __syncthreads();

    // Phase A: build h2 tile (416 x 16) in LDS
    for (int e = tid; e < CT * 16; e += 128) {
        const int mm = e & 15;
        const int r  = e >> 4;
        const float* hr = h1 + ((size_t)b * CT + r) * NN;
        float acc = 0.f;
        #pragma unroll
        for (int k = 0; k < KK; ++k)
            acc += sav2[mm * KK + k] * hr[sidx2[mm * KK + k]];
        const float xv = x[((size_t)b * CT + r) * NN + m0 + mm];
        h2s[e] = ALPHA * xv + (1.0f - ALPHA) * acc;
    }
    __syncthreads();

    // Phase B: WMMA projection
    const int wave = tid >> 5;
    const int lane = tid & 31;
    const int half = lane >> 4;   // 0: lanes 0-15, 1: lanes 16-31
    const int l16  = lane & 15;

    // Preload A fragments of W1 for both 16-row tiles, all 8 K-steps.
    // 16x4 f32 A layout: VGPR0 = K{0|2}, VGPR1 = K{1|3} split by lane half.
    v2f afrag[2][8];
    #pragma unroll
    for (int h = 0; h < 2; ++h) {
        const int o = h * 16 + l16;
        #pragma unroll
        for (int kk = 0; kk < 8; ++kk) {
            const int c = kk * 4 + half * 2;
            v2f a; a[0] = W1[o * CIN + c]; a[1] = W1[o * CIN + c + 1];
            afrag[h][kk] = a;
        }
    }

    for (int t = wave; t < TT; t += 4) {
        #pragma unroll
        for (int h = 0; h < 2; ++h) {
            // init accumulator with bias (C/D layout: VGPR r -> M = r + 8*half)
            v8f acc;
            #pragma unroll
            for (int r = 0; r < 8; ++r)
                acc[r] = b1[h * 16 + r + half * 8];

            #pragma unroll
            for (int kk = 0; kk < 8; ++kk) {
                const int c = kk * 4 + half * 2;
                v2f bf;
                bf[0] = h2s[((c    ) * TT + t) * 16 + l16];
                bf[1] = h2s[((c + 1) * TT + t) * 16 + l16];
                acc = __builtin_amdgcn_wmma_f32_16x16x4_f32(
                        false, afrag[h][kk], false, bf,
                        (short)0, acc, false, false);
            }

            #pragma unroll
            for (int r = 0; r < 8; ++r) {
                const int o = h * 16 + r + half * 8;
                out[(((size_t)b * COUT + o) * TT + t) * NN + m0 + l16] = acc[r];
            }
        }
    }
}

// ---------------------------------------------------------------------------
extern "C" void kernel_launch(void* const* d_in, const int* in_sizes, int n_in,
                              void* d_out, int out_size, void* d_ws, size_t ws_size,
                              hipStream_t stream) {
    const float* x   = (const float*)d_in[0];
    const float* adj = (const float*)d_in[1];
    const float* W1  = (const float*)d_in[2];
    const float* b1  = (const float*)d_in[3];
    const float* W2  = (const float*)d_in[4];
    const float* b2  = (const float*)d_in[5];
    const float* a   = (const float*)d_in[6];
    float* out = (float*)d_out;

    // workspace layout
    char* p = (char*)d_ws;
    auto alloc = [&](size_t bytes) { char* r = p; p += (bytes + 255) & ~size_t(255); return r; };
    float* u1    = (float*)alloc(CT * sizeof(float));
    float* u2    = (float*)alloc(CT * sizeof(float));
    float* c12   = (float*)alloc(2 * sizeof(float));
    int*   idx   = (int*)  alloc((size_t)NN * KK * sizeof(int));
    float* aval  = (float*)alloc((size_t)NN * KK * sizeof(float));
    float* left  = (float*)alloc((size_t)BB * NN * sizeof(float));
    float* right = (float*)alloc((size_t)BB * NN * sizeof(float));
    float* denom = (float*)alloc((size_t)BB * NN * sizeof(float));
    float* rmax  = (float*)alloc(BB * sizeof(float));
    float* Aval  = (float*)alloc((size_t)BB * NN * KK * sizeof(float));
    float* h1    = (float*)alloc((size_t)BB * CT * NN * sizeof(float));

    k_uvec<<<1, 512, 0, stream>>>(W2, b2, a, u1, u2, c12);
    k_adj_topk<<<NN, 256, 0, stream>>>(adj, idx, aval);
    k_leftright<<<dim3((NN + 255) / 256, BB), 256, 0, stream>>>(x, u1, u2, c12, left, right);
    k_denom<<<dim3((NN + 255) / 256, BB), 256, 0, stream>>>(left, right, denom, rmax);
    k_attn<<<(BB * NN * KK + 255) / 256, 256, 0, stream>>>(idx, aval, left, right, denom, rmax, Aval);
    k_prop1<<<dim3(NN, BB), 256, 0, stream>>>(x, idx, Aval, h1);
    k_prop2_proj<<<dim3(NN / 16, BB), 128, 0, stream>>>(x, h1, idx, Aval, W1, b1, out);
}